// CTCModel_13082470384198
// MI455X (gfx1250) — compile-verified
//
#include <hip/hip_runtime.h>

// CTC forward (Keras ctc_batch_cost semantics) for B=1024, T=256, C=128, L=64.
// One block per batch element; 128 threads = 4 wave32.
// CDNA5 path: 4-deep ring of GLOBAL_LOAD_ASYNC_TO_LDS_B32 row DMAs (ASYNCcnt),
// steady-state s_wait_asynccnt 2 so up to 3 rows of DMA lag behind compute.

namespace {
constexpr int   kB    = 1024;
constexpr int   kT    = 256;
constexpr int   kC    = 128;
constexpr int   kL    = 64;
constexpr int   kS    = 2 * kL + 1;   // 129 extended states
constexpr int   BLANK = kC - 1;       // 127
constexpr float NEGF  = -1e30f;
constexpr float EPSF  = 1e-7f;
constexpr int   kD    = 4;            // async ring depth

__device__ __forceinline__ float lae3(float a, float b, float c) {
  float m = fmaxf(a, fmaxf(b, c));
  return m + __logf(__expf(a - m) + __expf(b - m) + __expf(c - m));
}
__device__ __forceinline__ float lae2(float a, float b) {
  float m = fmaxf(a, b);
  return m + __logf(__expf(a - m) + __expf(b - m));
}
} // namespace

__global__ __launch_bounds__(128) void ctc_loss_kernel(
    const float* __restrict__ y_pred,    // [B, T, C]
    const int*   __restrict__ y_true,    // [B, L]
    const int*   __restrict__ label_len, // [B]
    float*       __restrict__ loss)      // [B]
{
  __shared__ float ylds[kD][kC];       // async DMA ring of class rows
  __shared__ float alpha[2][kS + 2];   // [0],[1] = permanent NEG padding
  __shared__ int   extsh[kS];
  __shared__ float wsum[4];

  const int b   = blockIdx.x;
  const int tid = threadIdx.x;
  const float* grow = y_pred + (size_t)b * kT * kC;

  // Extended label sequence: blank, l1, blank, ..., blank
  if (tid <= kL) extsh[2 * tid] = BLANK;                 // s = 0,2,...,128
  if (tid <  kL) extsh[2 * tid + 1] = y_true[b * kL + tid];
  for (int i = tid; i < kS + 2; i += 128) {
    alpha[0][i] = NEGF;
    alpha[1][i] = NEGF;
  }

  // --- CDNA5 async global->LDS row streaming (ASYNCcnt) ---
  auto issue_row = [&](int t) {
    unsigned int       ldsoff = (unsigned int)(unsigned long long)(uintptr_t)&ylds[t & (kD - 1)][tid];
    unsigned long long gaddr  = (unsigned long long)(uintptr_t)(grow + (size_t)t * kC + tid);
    asm volatile("global_load_async_to_lds_b32 %0, %1, off"
                 :
                 : "v"(ldsoff), "v"(gaddr)
                 : "memory");
  };
  // Wait so that row t (oldest outstanding) has landed; loads complete in order.
  auto wait_row = [&](int t) {
    if (t <= kT - kD + 1)  // rows t+1, t+2 may still be in flight
      asm volatile("s_wait_asynccnt 2" ::: "memory");
    else if (t == kT - 2)
      asm volatile("s_wait_asynccnt 1" ::: "memory");
    else
      asm volatile("s_wait_asynccnt 0" ::: "memory");
  };

  issue_row(0);
  issue_row(1);
  issue_row(2);

  __syncthreads();  // extsh/alpha init visible

  const int  myext = extsh[tid];
  const bool allow = (tid >= 2) && (myext != BLANK) && (myext != extsh[tid - 2]);

  float* Acur = alpha[0];
  float* Anew = alpha[1];

  // Row sum over 128 classes: wave32 butterfly + 4-wave LDS combine.
  // Leaves threads past a barrier with `sum` valid; one barrier inside.
  auto row_sum = [&](int buf) -> float {
    float v = ylds[buf][tid];
#pragma unroll
    for (int off = 16; off > 0; off >>= 1) v += __shfl_xor(v, off, 32);
    if ((tid & 31) == 0) wsum[tid >> 5] = v;
    __syncthreads();
    return wsum[0] + wsum[1] + wsum[2] + wsum[3];
  };

  // ---- t = 0 (peeled): alpha0[s] = emit0[s] for s < 2, else NEG (already NEG)
  {
    wait_row(0);
    __syncthreads();          // all waves' row-0 lanes landed
    issue_row(3);
    const float sum  = row_sum(0);
    const float lsum = __logf(sum);
    if (tid < 2)
      Acur[tid + 2] = __logf(ylds[0][myext] + EPSF * sum) - lsum;
  }

  // ---- t = 1 .. T-1: hot DP loop, no t==0 branch
  for (int t = 1; t < kT; ++t) {
    const int buf = t & (kD - 1);
    wait_row(t);
    __syncthreads();                       // row t landed everywhere; prev alpha visible
    if (t + 3 < kT) issue_row(t + 3);      // keep the ring full

    const float sum  = row_sum(buf);
    const float lsum = __logf(sum);
    const float esum = EPSF * sum;

    // emit[s] = log(y[ext[s]]/sum + eps) = log(y[ext[s]] + eps*sum) - log(sum)
    const float e  = __logf(ylds[buf][myext] + esum) - lsum;
    const float a  = Acur[tid + 2];
    const float a1 = Acur[tid + 1];
    const float a2 = allow ? Acur[tid] : NEGF;
    Anew[tid + 2] = lae3(a, a1, a2) + e;

    if (tid == 0) {  // s = 128: trailing blank, skip transition never allowed
      const float e2 = __logf(ylds[buf][BLANK] + esum) - lsum;
      Anew[kS + 1] = lae2(Acur[kS + 1], Acur[kS]) + e2;
    }

    float* tmp = Acur; Acur = Anew; Anew = tmp;
  }
  __syncthreads();

  if (tid == 0) {
    int lab = label_len[b];
    lab = lab < 1 ? 1 : (lab > kL ? kL : lab);
    const int end = 2 * lab;  // final states end-1, end (alpha index +2 pad)
    loss[b] = -lae2(Acur[end + 1], Acur[end + 2]);
  }
}

extern "C" void kernel_launch(void* const* d_in, const int* in_sizes, int n_in,
                              void* d_out, int out_size, void* d_ws, size_t ws_size,
                              hipStream_t stream) {
  (void)in_sizes; (void)n_in; (void)d_ws; (void)ws_size; (void)out_size;
  const float* y_pred    = (const float*)d_in[0];
  const int*   y_true    = (const int*)d_in[1];
  const int*   label_len = (const int*)d_in[2];
  float*       out       = (float*)d_out;

  ctc_loss_kernel<<<dim3(kB), dim3(128), 0, stream>>>(y_pred, y_true, label_len, out);
}